// MAGNN_9689446220160
// MI455X (gfx1250) — compile-verified
//
#include <hip/hip_runtime.h>
#include <math.h>

#define IN_F    128
#define HID     64
#define NNODES  100000
#define NEDGES  1000000

typedef __attribute__((ext_vector_type(2))) float v2f;
typedef __attribute__((ext_vector_type(8))) float v8f;

// ---------------------------------------------------------------------------
// Kernel 0: initialize out = bias (broadcast), m = -inf, denom = 0
// ---------------------------------------------------------------------------
__global__ void k_init(float* __restrict__ out, float* __restrict__ m,
                       float* __restrict__ denom, const float* __restrict__ bias) {
    int tid = blockIdx.x * blockDim.x + threadIdx.x;
    if (tid < NNODES * HID) out[tid] = bias[tid & (HID - 1)];
    if (tid < NNODES) { m[tid] = -__builtin_inff(); denom[tid] = 0.0f; }
}

// ---------------------------------------------------------------------------
// Kernel 1: H[z] = feat[z] @ W_feat + b_feat   via V_WMMA_F32_16X16X4_F32
// One wave computes a 16x64 tile; 8 waves / block; blockIdx.z selects feat.
// W_feat staged in LDS with K-pairs interleaved: sW2[k/2][col][2], so each
// B fragment {W[ka][col], W[ka+1][col]} is ONE aligned ds_load_b64 straight
// into an even VGPR pair (no repacking movs).
// ---------------------------------------------------------------------------
__global__ void __launch_bounds__(256)
k_gemm(const float* __restrict__ feat0, const float* __restrict__ feat1,
       const float* __restrict__ feat2, const float* __restrict__ W,
       const float* __restrict__ b, float* __restrict__ H) {
    __shared__ float sW2[IN_F * HID];                       // 32 KB, paired layout
    const float* feat = (blockIdx.z == 0) ? feat0 : ((blockIdx.z == 1) ? feat1 : feat2);
    float* Hout = H + (size_t)blockIdx.z * NNODES * HID;

    // stage W with pair interleave: sW2[(k>>1)*128 + n*2 + (k&1)] = W[k][n]
    for (int i = threadIdx.x; i < IN_F * HID; i += 256) {
        int k = i / HID, n = i - k * HID;
        sW2[(k >> 1) * (2 * HID) + n * 2 + (k & 1)] = W[i];
    }
    __syncthreads();

    const int wave = threadIdx.x >> 5;
    const int lane = threadIdx.x & 31;
    const int nTiles = NNODES / 16;                         // 6250
    const int tile = blockIdx.x * 8 + wave;
    if (tile >= nTiles) return;                             // wave-uniform exit

    const int row0 = tile * 16;
    const int half = lane >> 4;                             // 0 or 1
    const int l15  = lane & 15;
    const float* arow = feat + (size_t)(row0 + l15) * IN_F;
    // base pointer into paired-W for this lane's column slot
    const float* wbase = &sW2[half * (2 * HID) + l15 * 2];

    v8f acc0 = {}, acc1 = {}, acc2 = {}, acc3 = {};

    #pragma unroll 4
    for (int k0 = 0; k0 < IN_F; k0 += 4) {
        const int ka = k0 + 2 * half;
        // A fragment: contiguous global_load_b64 (row row0+l15, K=ka..ka+1)
        v2f a = *(const v2f*)(arow + ka);
        // B fragments: one ds_load_b64 each; pair row (k0>>1)+half
        const float* wp = wbase + (k0 >> 1) * (2 * HID);
        v2f b0 = *(const v2f*)(wp);
        v2f b1 = *(const v2f*)(wp + 32);
        v2f b2 = *(const v2f*)(wp + 64);
        v2f b3 = *(const v2f*)(wp + 96);
        acc0 = __builtin_amdgcn_wmma_f32_16x16x4_f32(false, a, false, b0, (short)0, acc0, false, false);
        acc1 = __builtin_amdgcn_wmma_f32_16x16x4_f32(false, a, false, b1, (short)0, acc1, false, false);
        acc2 = __builtin_amdgcn_wmma_f32_16x16x4_f32(false, a, false, b2, (short)0, acc2, false, false);
        acc3 = __builtin_amdgcn_wmma_f32_16x16x4_f32(false, a, false, b3, (short)0, acc3, false, false);
    }

    v8f accs[4] = {acc0, acc1, acc2, acc3};
    #pragma unroll
    for (int n = 0; n < 4; n++) {
        const int col = n * 16 + l15;
        const float bv = b[col];
        #pragma unroll
        for (int v = 0; v < 8; v++) {
            const int r = row0 + v + 8 * half;              // C layout: lanes 16-31 -> M+8
            Hout[(size_t)r * HID + col] = accs[n][v] + bv;
        }
    }
}

// ---------------------------------------------------------------------------
// float atomic-max via signed/unsigned int ordering trick (init must be -inf)
// ---------------------------------------------------------------------------
__device__ __forceinline__ void atomicMaxF(float* addr, float v) {
    if (v >= 0.0f) atomicMax((int*)addr, __float_as_int(v));
    else           atomicMin((unsigned int*)addr, (unsigned int)__float_as_int(v));
}

// ---------------------------------------------------------------------------
// Kernel 2: per-edge attention logit e = tanh([h0, enc] . W_att + b_att)
//           + atomic segment-max into m[edge0]
// ---------------------------------------------------------------------------
__global__ void k_score(const float* __restrict__ H, const int* __restrict__ e0,
                        const int* __restrict__ e1, const int* __restrict__ e2,
                        const float* __restrict__ W_att, const float* __restrict__ b_att,
                        float* __restrict__ e_arr, float* __restrict__ m) {
    int i = blockIdx.x * blockDim.x + threadIdx.x;
    if (i >= NEDGES) return;
    const int d = e0[i];
    const float* h0 = H + (size_t)d * HID;
    const float* h1 = H + (size_t)NNODES * HID     + (size_t)e1[i] * HID;
    const float* h2 = H + (size_t)2 * NNODES * HID + (size_t)e2[i] * HID;
    float acc = b_att[0];
    #pragma unroll
    for (int j = 0; j < HID; j += 4) {
        float4 a0 = *(const float4*)(h0 + j);
        float4 a1 = *(const float4*)(h1 + j);
        float4 a2 = *(const float4*)(h2 + j);
        float4 wa = *(const float4*)(W_att + j);
        float4 wb = *(const float4*)(W_att + HID + j);
        const float k = 1.0f / 3.0f;
        float ex_ = (a0.x + a1.x + a2.x) * k;
        float ey_ = (a0.y + a1.y + a2.y) * k;
        float ez_ = (a0.z + a1.z + a2.z) * k;
        float ew_ = (a0.w + a1.w + a2.w) * k;
        acc += a0.x * wa.x + a0.y * wa.y + a0.z * wa.z + a0.w * wa.w;
        acc += ex_  * wb.x + ey_  * wb.y + ez_  * wb.z + ew_  * wb.w;
    }
    float ev = tanhf(acc);
    e_arr[i] = ev;
    atomicMaxF(&m[d], ev);
}

// ---------------------------------------------------------------------------
// Kernel 3: ex = exp(e - m[edge0]) (in place) + atomic segment-sum into denom
// ---------------------------------------------------------------------------
__global__ void k_exp(float* __restrict__ e_arr, const int* __restrict__ e0,
                      const float* __restrict__ m, float* __restrict__ denom) {
    int i = blockIdx.x * blockDim.x + threadIdx.x;
    if (i >= NEDGES) return;
    const int d = e0[i];
    float v = __expf(e_arr[i] - m[d]);
    e_arr[i] = v;
    atomicAdd(&denom[d], v);
}

// ---------------------------------------------------------------------------
// Kernel 4: h_prime[edge0] += (ex/denom[edge0]) * enc   (scatter-add, L2-resident)
// ---------------------------------------------------------------------------
__global__ void k_agg(const float* __restrict__ H, const int* __restrict__ e0,
                      const int* __restrict__ e1, const int* __restrict__ e2,
                      const float* __restrict__ e_arr, const float* __restrict__ denom,
                      float* __restrict__ out) {
    int i = blockIdx.x * blockDim.x + threadIdx.x;
    if (i >= NEDGES) return;
    const int d = e0[i];
    const float w = e_arr[i] / denom[d];
    const float* h0 = H + (size_t)d * HID;
    const float* h1 = H + (size_t)NNODES * HID     + (size_t)e1[i] * HID;
    const float* h2 = H + (size_t)2 * NNODES * HID + (size_t)e2[i] * HID;
    float* o = out + (size_t)d * HID;
    const float k = w * (1.0f / 3.0f);
    #pragma unroll
    for (int j = 0; j < HID; j += 4) {
        float4 a0 = *(const float4*)(h0 + j);
        float4 a1 = *(const float4*)(h1 + j);
        float4 a2 = *(const float4*)(h2 + j);
        atomicAdd(o + j + 0, k * (a0.x + a1.x + a2.x));
        atomicAdd(o + j + 1, k * (a0.y + a1.y + a2.y));
        atomicAdd(o + j + 2, k * (a0.z + a1.z + a2.z));
        atomicAdd(o + j + 3, k * (a0.w + a1.w + a2.w));
    }
}

// ---------------------------------------------------------------------------
extern "C" void kernel_launch(void* const* d_in, const int* in_sizes, int n_in,
                              void* d_out, int out_size, void* d_ws, size_t ws_size,
                              hipStream_t stream) {
    const float* feat0 = (const float*)d_in[0];
    const float* feat1 = (const float*)d_in[1];
    const float* feat2 = (const float*)d_in[2];
    const int*   edge0 = (const int*)d_in[3];
    const int*   edge1 = (const int*)d_in[4];
    const int*   edge2 = (const int*)d_in[5];
    const float* W_feat = (const float*)d_in[6];
    const float* b_feat = (const float*)d_in[7];
    const float* W_att  = (const float*)d_in[8];
    const float* b_att  = (const float*)d_in[9];
    const float* bias   = (const float*)d_in[10];
    float* out = (float*)d_out;

    // workspace layout (floats): H[3*N*HID] | e[E] | m[N] | denom[N]  (~82 MB)
    float* ws    = (float*)d_ws;
    float* H     = ws;
    float* e_arr = H + (size_t)3 * NNODES * HID;
    float* m     = e_arr + NEDGES;
    float* denom = m + NNODES;

    // 0) init
    {
        int n = NNODES * HID;
        k_init<<<(n + 255) / 256, 256, 0, stream>>>(out, m, denom, bias);
    }
    // 1) WMMA GEMM: H[z] = feat[z] @ W_feat + b_feat
    {
        int nTiles = NNODES / 16;                 // 6250
        dim3 grid((nTiles + 7) / 8, 1, 3);
        k_gemm<<<grid, 256, 0, stream>>>(feat0, feat1, feat2, W_feat, b_feat, H);
    }
    // 2) edge logits + segment max
    k_score<<<(NEDGES + 255) / 256, 256, 0, stream>>>(H, edge0, edge1, edge2,
                                                      W_att, b_att, e_arr, m);
    // 3) exp + segment sum
    k_exp<<<(NEDGES + 255) / 256, 256, 0, stream>>>(e_arr, edge0, m, denom);
    // 4) weighted aggregation scatter-add
    k_agg<<<(NEDGES + 255) / 256, 256, 0, stream>>>(H, edge0, edge1, edge2,
                                                    e_arr, denom, out);
}